// minSAD_transforms_elisheva_76020921139281
// MI455X (gfx1250) — compile-verified
//
#include <hip/hip_runtime.h>
#include <hip/hip_bf16.h>
#include <math.h>

typedef __attribute__((ext_vector_type(2))) float v2f;
typedef __attribute__((ext_vector_type(8))) float v8f;

#define IMG_H 512
#define IMG_W 512
#define PI_F 3.14159265358979323846f

// ---------------------------------------------------------------------------
// Kernel 1: one workgroup per (candidate, image) pair. Computes
// SAD[cand][bt] = mean |bilinear_warp(img) - center_crop(ref)| over the crop.
//
// - Row-tiled: each block iteration processes one crop row; each thread owns
//   a fixed column pair, so the column term of the affine map is loop
//   invariant (1 cvt + 2 fma per pixel, no divergent wrap logic).
// - Boundary masks folded into bilinear weights -> unconditional clamped
//   gathers, EXEC stays all-ones (WMMA requirement).
// - Abs-diffs accumulated on the matrix pipe via V_WMMA_F32_16X16X4_F32
//   (A = 64 abs-diffs, B = ones, C += row sums), freeing the VALU pipe for
//   the gather/address math.
// ---------------------------------------------------------------------------
__global__ __launch_bounds__(256) void sad_kernel(
    const float* __restrict__ img, const float* __restrict__ ref,
    const float* __restrict__ shv, const float* __restrict__ swv,
    const float* __restrict__ rvv, const float* __restrict__ svv,
    int n_h, int n_w, int n_r, int n_s, int BT, float* __restrict__ out)
{
    const int cand = blockIdx.x;
    const int bt   = blockIdx.y;
    const int tid  = threadIdx.x;

    // unravel candidate index (ih, iw, ir, isc), 'ij' meshgrid order
    int tmp = cand;
    const int isc = tmp % n_s; tmp /= n_s;
    const int ir  = tmp % n_r; tmp /= n_r;
    const int iw  = tmp % n_w;
    const int ih  = tmp / n_w;

    // ---- crop sizes (reference computes these host-side in float32) ----
    const float k_deg = PI_F / 180.0f;
    float max_sh = 0.0f, max_sw = 0.0f;
    for (int i = 0; i < n_h; ++i) max_sh = fmaxf(max_sh, fabsf(shv[i]));
    for (int i = 0; i < n_w; ++i) max_sw = fmaxf(max_sw, fabsf(swv[i]));
    float max_rr = rvv[0] * k_deg;
    for (int i = 1; i < n_r; ++i) max_rr = fmaxf(max_rr, rvv[i] * k_deg);
    float max_sc = svv[0];
    for (int i = 1; i < n_s; ++i) max_sc = fmaxf(max_sc, svv[i]);
    const float tanmag = fabsf(tanf(max_rr));
    const int ch = (int)((float)IMG_H - (2.0f * (max_sh + 0.5f * (float)IMG_H * tanmag) * max_sc + 10.0f));
    const int cw = (int)((float)IMG_W - (2.0f * (max_sw + 0.5f * (float)IMG_W * tanmag) * max_sc + 10.0f));
    const int y1 = (IMG_H - ch) / 2;
    const int x1 = (IMG_W - cw) / 2;

    // ---- affine sampling coefficients: ix = axx*u + axy*v + ax0, etc. ----
    const float s_h = shv[ih];
    const float s_w = swv[iw];
    const float rot = rvv[ir] * k_deg;
    const float inv = 1.0f / svv[isc];
    const float c = cosf(rot);
    const float s = sinf(rot);
    const float cC  = (float)(IMG_W - 1) * 0.5f;               // == (H-1)/2, square
    const float kx  = (float)IMG_W / (float)(IMG_W - 1);       // == H/(H-1)
    const float axx =  kx * c * inv;
    const float axy =  kx * s * inv;
    const float ax0 =  kx * inv * ( c * (s_w - cC) + s * (s_h - cC)) + cC;
    const float ayx = -kx * s * inv;
    const float ayy =  kx * c * inv;
    const float ay0 =  kx * inv * (-s * (s_w - cC) + c * (s_h - cC)) + cC;

    const float* __restrict__ imgp = img + (size_t)bt * (IMG_H * IMG_W);
    const float* __restrict__ refp = ref + (size_t)bt * (IMG_H * IMG_W);

    // ---- per-thread column pair (loop invariant) ----
    const int   wc0 = tid * 2;           // crop columns wc0, wc0+1 (masked if >= cw)
    const float u0  = (float)(wc0 + x1);
    const float u1  = (float)(wc0 + 1 + x1);
    const float bx0 = fmaf(axx, u0, ax0);    // ix(row v) = fma(axy, v, bx)
    const float by0 = fmaf(ayx, u0, ay0);    // iy(row v) = fma(ayy, v, by)
    const float bx1 = fmaf(axx, u1, ax0);
    const float by1 = fmaf(ayx, u1, ay0);
    const float valid0 = (wc0     < cw) ? 1.0f : 0.0f;
    const float valid1 = (wc0 + 1 < cw) ? 1.0f : 0.0f;

    // bilinear sample + abs-diff; masks folded into weights, loads unconditional
    auto pixdiff = [&](float bx, float by, float vf, int refidx) -> float {
        const float ixf = fmaf(axy, vf, bx);
        const float iyf = fmaf(ayy, vf, by);
        const float fx = floorf(ixf), fy = floorf(iyf);
        const int x0 = (int)fx, y0 = (int)fy;
        const int x1i = x0 + 1, y1i = y0 + 1;
        const float mx0 = ((unsigned)x0  < (unsigned)IMG_W) ? 1.0f : 0.0f;
        const float mx1 = ((unsigned)x1i < (unsigned)IMG_W) ? 1.0f : 0.0f;
        const float my0 = ((unsigned)y0  < (unsigned)IMG_H) ? 1.0f : 0.0f;
        const float my1 = ((unsigned)y1i < (unsigned)IMG_H) ? 1.0f : 0.0f;
        const float wx1 = ixf - fx;
        const float wy1 = iyf - fy;
        const float wx0m = (1.0f - wx1) * mx0;
        const float wx1m = wx1 * mx1;
        const float wy0m = (1.0f - wy1) * my0;
        const float wy1m = wy1 * my1;
        const int xc0 = min(max(x0, 0), IMG_W - 1);
        const int xc1 = min(max(x1i, 0), IMG_W - 1);
        const int row0 = min(max(y0, 0), IMG_H - 1) << 9;     // *IMG_W
        const int row1 = min(max(y1i, 0), IMG_H - 1) << 9;
        const float t00 = imgp[row0 + xc0];
        const float t10 = imgp[row0 + xc1];
        const float t01 = imgp[row1 + xc0];
        const float t11 = imgp[row1 + xc1];
        const float rpix = refp[refidx];
        const float warped = fmaf(wy0m, fmaf(t00, wx0m, t10 * wx1m),
                                  wy1m * fmaf(t01, wx0m, t11 * wx1m));
        return fabsf(warped - rpix);
    };

    const v2f ones = {1.0f, 1.0f};
    v8f acc = {};

    int refidx = y1 * IMG_W + (wc0 + x1);    // advances by IMG_W per row

    #pragma unroll 2
    for (int hc = 0; hc < ch; ++hc) {
        const float vf = (float)(hc + y1);   // shared by both pixels
        const float d0 = pixdiff(bx0, by0, vf, refidx)     * valid0;
        const float d1 = pixdiff(bx1, by1, vf, refidx + 1) * valid1;

        v2f a; a[0] = d0; a[1] = d1;
        // acc[m][n] += sum_k A[m][k] * 1  (row-sum accumulation on matrix pipe)
        acc = __builtin_amdgcn_wmma_f32_16x16x4_f32(
            /*neg_a=*/false, a, /*neg_b=*/false, ones,
            /*c_mod=*/(short)0, acc, /*reuse_a=*/false, /*reuse_b=*/false);

        refidx += IMG_W;
    }

    // Per-lane sum of accumulator rows. Summed over all 32 lanes this equals
    // 16 * (true total), since B==ones replicates each row-sum across 16 cols.
    float lane_sum = acc[0] + acc[1] + acc[2] + acc[3] + acc[4] + acc[5] + acc[6] + acc[7];

    __shared__ float red[256];
    red[tid] = lane_sum;
    __syncthreads();
    #pragma unroll
    for (int off = 128; off > 0; off >>= 1) {
        if (tid < off) red[tid] += red[tid + off];
        __syncthreads();
    }
    if (tid == 0) {
        const int N = ch * cw;
        out[4 * BT + cand * BT + bt] = (red[0] * (1.0f / 16.0f)) / (float)N;
    }
}

// ---------------------------------------------------------------------------
// Kernel 2: per-image argmin over candidates + parabola subpixel refinement.
// ---------------------------------------------------------------------------
__device__ __forceinline__ float subpix(const float* vec, int idx, float y1, float y2, float y3) {
    float denom = y1 - 2.0f * y2 + y3;
    if (fabsf(denom) < 1e-12f) denom = 1e-12f;
    float delta = 0.5f * (y1 - y3) / denom;
    delta = fminf(fmaxf(delta, -1.0f), 1.0f);
    const float step = vec[1] - vec[0];
    return vec[idx] + delta * step;
}

__global__ void argmin_kernel(const float* __restrict__ shv, const float* __restrict__ swv,
                              const float* __restrict__ rvv, const float* __restrict__ svv,
                              int n_h, int n_w, int n_r, int n_s, int BT, float* out)
{
    const int t = threadIdx.x;
    if (t >= BT) return;
    const float* SAD = out + 4 * BT;     // SAD[cand*BT + bt], written by sad_kernel
    const int ncand = n_h * n_w * n_r * n_s;

    float best = SAD[t];
    int bidx = 0;
    for (int cc = 1; cc < ncand; ++cc) {
        const float v = SAD[cc * BT + t];
        if (v < best) { best = v; bidx = cc; }   // strict < keeps first occurrence
    }
    int tmp = bidx;
    int isc = tmp % n_s; tmp /= n_s;
    int ir  = tmp % n_r; tmp /= n_r;
    int iw  = tmp % n_w;
    int ih  = tmp / n_w;
    ih  = min(max(ih, 1), n_h - 2);
    iw  = min(max(iw, 1), n_w - 2);
    ir  = min(max(ir, 1), n_r - 2);
    isc = min(max(isc, 1), n_s - 2);

    auto sad4 = [&](int a, int b, int c, int d) -> float {
        const int cc = ((a * n_w + b) * n_r + c) * n_s + d;
        return SAD[cc * BT + t];
    };
    const float y2 = sad4(ih, iw, ir, isc);
    out[0 * BT + t] = subpix(shv, ih,  sad4(ih - 1, iw, ir, isc), y2, sad4(ih + 1, iw, ir, isc));
    out[1 * BT + t] = subpix(swv, iw,  sad4(ih, iw - 1, ir, isc), y2, sad4(ih, iw + 1, ir, isc));
    out[2 * BT + t] = subpix(rvv, ir,  sad4(ih, iw, ir - 1, isc), y2, sad4(ih, iw, ir + 1, isc));
    out[3 * BT + t] = subpix(svv, isc, sad4(ih, iw, ir, isc - 1), y2, sad4(ih, iw, ir, isc + 1));
}

extern "C" void kernel_launch(void* const* d_in, const int* in_sizes, int n_in,
                              void* d_out, int out_size, void* d_ws, size_t ws_size,
                              hipStream_t stream) {
    const float* img = (const float*)d_in[0];
    const float* ref = (const float*)d_in[1];
    const float* shv = (const float*)d_in[2];
    const float* swv = (const float*)d_in[3];
    const float* rvv = (const float*)d_in[4];
    const float* svv = (const float*)d_in[5];

    const int n_h = in_sizes[2];
    const int n_w = in_sizes[3];
    const int n_r = in_sizes[4];
    const int n_s = in_sizes[5];
    const int BT  = in_sizes[0] / (IMG_H * IMG_W);
    const int ncand = n_h * n_w * n_r * n_s;

    dim3 grid(ncand, BT);
    sad_kernel<<<grid, 256, 0, stream>>>(img, ref, shv, swv, rvv, svv,
                                         n_h, n_w, n_r, n_s, BT, (float*)d_out);
    argmin_kernel<<<1, 32, 0, stream>>>(shv, swv, rvv, svv,
                                        n_h, n_w, n_r, n_s, BT, (float*)d_out);
}